// ImprovedQLayer_90228672955300
// MI455X (gfx1250) — compile-verified
//
#include <hip/hip_runtime.h>
#include <math.h>

typedef __attribute__((ext_vector_type(2))) float v2f;
typedef __attribute__((ext_vector_type(8))) float v8f;

#define PI_F 3.14159265358979f

// Fast device trig: arguments here are bounded (|a| <= pi), so the native
// CDNA5 TRANS ops (v_sin_f32 / v_cos_f32 / v_exp_f32, ~1-2 ulp) are accurate
// and avoid the huge libm slow-path expansions (Payne-Hanek etc.).
__device__ __forceinline__ float fsin(float x) { return __sinf(x); }
__device__ __forceinline__ float fcos(float x) { return __cosf(x); }
__device__ __forceinline__ float ftanh(float x) {
    float ax = fabsf(x);
    float e  = __expf(2.0f * ax);                      // v_exp_f32
    float r  = 1.0f - 2.0f * __builtin_amdgcn_rcpf(e + 1.0f);  // -> 1 as e->inf
    return copysignf(r, x);
}

// ---------------------------------------------------------------------------
// Kernel 1: build the variational unitary U_var (16x16 complex) from q_weights
// and store it real-ified as the WMMA B operand BM[32][32] (row-major, K x N):
//   out[n] = sum_K BM[K][n] * v[K]   with v[2k]=Re in_k, v[2k+1]=Im in_k
// Thread t owns column t of U (input basis state t). All loops unrolled so the
// column stays in registers (all indices compile-time constant).
// ---------------------------------------------------------------------------
__global__ void build_umat(const float* __restrict__ qw, float* __restrict__ BM) {
    int t = threadIdx.x;
    if (t < 16) {
        float cr[16], ci[16];
        #pragma unroll
        for (int k = 0; k < 16; ++k) { cr[k] = (k == t) ? 1.f : 0.f; ci[k] = 0.f; }

        #pragma unroll
        for (int l = 0; l < 3; ++l) {
            // Rot(phi, theta, omega) = RZ(omega) RY(theta) RZ(phi) on each wire
            #pragma unroll
            for (int i = 0; i < 4; ++i) {
                float phi = qw[(l*4 + i)*3 + 0];
                float th  = qw[(l*4 + i)*3 + 1];
                float om  = qw[(l*4 + i)*3 + 2];
                float c = fcos(0.5f*th), s = fsin(0.5f*th);
                float apo = 0.5f*(phi + om);
                float amo = 0.5f*(phi - om);
                // g00 =  c e^{-i apo}; g01 = -s e^{+i amo}
                // g10 =  s e^{-i amo}; g11 =  c e^{+i apo}
                float cpo = fcos(apo), spo = fsin(apo);
                float cmo = fcos(amo), smo = fsin(amo);
                float g00r =  c*cpo, g00i = -c*spo;
                float g11r =  c*cpo, g11i =  c*spo;
                float g01r = -s*cmo, g01i = -s*smo;
                float g10r =  s*cmo, g10i = -s*smo;
                int mask = 8 >> i;
                #pragma unroll
                for (int r0 = 0; r0 < 16; ++r0) {
                    if (r0 & mask) continue;
                    int r1 = r0 | mask;
                    float x0r = cr[r0], x0i = ci[r0];
                    float x1r = cr[r1], x1i = ci[r1];
                    cr[r0] = g00r*x0r - g00i*x0i + g01r*x1r - g01i*x1i;
                    ci[r0] = g00r*x0i + g00i*x0r + g01r*x1i + g01i*x1r;
                    cr[r1] = g10r*x0r - g10i*x0i + g11r*x1r - g11i*x1i;
                    ci[r1] = g10r*x0i + g10i*x0r + g11r*x1i + g11i*x1r;
                }
            }
            // CNOT ring: control i, target (i+1)%4; new[k] = old[k^mt] if k&mc
            #pragma unroll
            for (int i = 0; i < 4; ++i) {
                int mc = 8 >> i, mt = 8 >> ((i + 1) & 3);
                float nr[16], ni[16];
                #pragma unroll
                for (int k = 0; k < 16; ++k) {
                    int src = (k & mc) ? (k ^ mt) : k;
                    nr[k] = cr[src]; ni[k] = ci[src];
                }
                #pragma unroll
                for (int k = 0; k < 16; ++k) { cr[k] = nr[k]; ci[k] = ni[k]; }
            }
        }
        // Column t of U => K rows 2t, 2t+1 of BM (BM = M^T of the real 32x32 op)
        #pragma unroll
        for (int kp = 0; kp < 16; ++kp) {
            BM[(2*t    )*32 + 2*kp    ] =  cr[kp];
            BM[(2*t    )*32 + 2*kp + 1] =  ci[kp];
            BM[(2*t + 1)*32 + 2*kp    ] = -ci[kp];
            BM[(2*t + 1)*32 + 2*kp + 1] =  cr[kp];
        }
    }
}

// ---------------------------------------------------------------------------
// Kernel 2: per wave, 16 samples. Encoding -> 16x v_wmma_f32_16x16x4_f32
// (2 N-halves x 8 K-steps) -> |amp|^2 sign-reduction -> [B,4] output.
// A-fragment for lane half hi, step st covers K = {4st+2hi, 4st+2hi+1}, i.e.
// amplitude k = 2*st + hi => each lane computes ONLY its 8 needed amplitudes,
// with all private-array indices compile-time constant (no cndmask cascades).
// ---------------------------------------------------------------------------
__global__ __launch_bounds__(256) void qlayer_main(const float* __restrict__ x,
                                                   const float* __restrict__ BM,
                                                   float* __restrict__ out, int B) {
    __shared__ float lds[8 * 32 * 17];   // per-wave 32 x 16 result tile, stride 17

    const int lane = threadIdx.x & 31;
    const int wv   = threadIdx.x >> 5;
    const int hi   = lane >> 4;          // lane half (K sub-group of A/B frags)
    const int m    = lane & 15;          // sample-in-tile / N index
    const int tile = blockIdx.x * 8 + wv;
    const int s    = tile * 16 + m;      // global sample id

    float4 xv = make_float4(0.f, 0.f, 0.f, 0.f);
    if (s < B) xv = *(const float4*)(x + 4 * (size_t)s);

    // ---- per-qubit encoded states: RZ(a) RY(2a)|0> = [e^{-ia/2}cos a, e^{ia/2}sin a]
    float pr[4][2], pim[4][2];
    {
        float xs[4] = {xv.x, xv.y, xv.z, xv.w};
        #pragma unroll
        for (int i = 0; i < 4; ++i) {
            float a  = ftanh(xs[i]) * PI_F;
            float ca = fcos(a), sa = fsin(a);
            float ch = fcos(0.5f*a), sh = fsin(0.5f*a);
            pr[i][0] = ch*ca;  pim[i][0] = -sh*ca;
            pr[i][1] = ch*sa;  pim[i][1] =  sh*sa;
        }
    }

    // ---- pair products; q23 held as named scalars so the ONLY lane-variant
    //      choice is a value-level ternary (single v_cndmask each)
    float q01r[4], q01i[4];
    #pragma unroll
    for (int j = 0; j < 4; ++j) {
        int b0 = j >> 1, b1 = j & 1;
        q01r[j] = pr[0][b0]*pr[1][b1] - pim[0][b0]*pim[1][b1];
        q01i[j] = pr[0][b0]*pim[1][b1] + pim[0][b0]*pr[1][b1];
    }
    float q23r0 = pr[2][0]*pr[3][0] - pim[2][0]*pim[3][0];
    float q23i0 = pr[2][0]*pim[3][0] + pim[2][0]*pr[3][0];
    float q23r1 = pr[2][0]*pr[3][1] - pim[2][0]*pim[3][1];
    float q23i1 = pr[2][0]*pim[3][1] + pim[2][0]*pr[3][1];
    float q23r2 = pr[2][1]*pr[3][0] - pim[2][1]*pim[3][0];
    float q23i2 = pr[2][1]*pim[3][0] + pim[2][1]*pr[3][0];
    float q23r3 = pr[2][1]*pr[3][1] - pim[2][1]*pim[3][1];
    float q23i3 = pr[2][1]*pim[3][1] + pim[2][1]*pr[3][1];

    // jl = hi (even st) or 2+hi (odd st): two scalar selects on hi
    float qa_r = hi ? q23r1 : q23r0;   // jl = hi
    float qa_i = hi ? q23i1 : q23i0;
    float qb_r = hi ? q23r3 : q23r2;   // jl = 2 + hi
    float qb_i = hi ? q23i3 : q23i2;

    // ---- A fragments: afrag[st] = (Re, Im) of amp_{2*st+hi}; jh = st>>1 const
    v2f afrag[8];
    #pragma unroll
    for (int st = 0; st < 8; ++st) {
        int jh = st >> 1;
        float sr = (st & 1) ? qb_r : qa_r;
        float si = (st & 1) ? qb_i : qa_i;
        v2f A;
        A.x = q01r[jh]*sr - q01i[jh]*si;
        A.y = q01r[jh]*si + q01i[jh]*sr;
        afrag[st] = A;
    }

    // ---- preload B fragments (ISA layout: VGPR0 K={0,2}, VGPR1 K={1,3}) ----
    v2f bfrag[2][8];
    #pragma unroll
    for (int h = 0; h < 2; ++h) {
        int n = m + 16*h;
        #pragma unroll
        for (int st = 0; st < 8; ++st) {
            int kb = 4*st + 2*hi;
            v2f b; b.x = BM[kb*32 + n]; b.y = BM[(kb + 1)*32 + n];
            bfrag[h][st] = b;
        }
    }

    float* res = &lds[wv * (32 * 17)];

    // ---- W[16 samples][32 outs] = V @ BM via 16x v_wmma_f32_16x16x4_f32 ----
    #pragma unroll
    for (int h = 0; h < 2; ++h) {
        v8f acc = {};
        #pragma unroll
        for (int st = 0; st < 8; ++st) {
            acc = __builtin_amdgcn_wmma_f32_16x16x4_f32(
                      false, afrag[st], false, bfrag[h][st], (short)0, acc,
                      false, false);
        }
        // D layout: lane holds (M = r + 8*hi, N = m); store transposed [n][m']
        int n = m + 16*h;
        #pragma unroll
        for (int r = 0; r < 8; ++r)
            res[n*17 + 8*hi + r] = acc[r];
    }

    __syncthreads();

    // ---- ev_i = sum_k (+/-)|amp_k|^2 ; lane pair (l, l+16) shares sample m ----
    float ev[4] = {0.f, 0.f, 0.f, 0.f};
    #pragma unroll
    for (int k2 = 0; k2 < 8; ++k2) {
        int k = hi*8 + k2;                  // bit3 = hi (runtime), bits2..0 = k2
        float re = res[(2*k    )*17 + m];   // LDS: runtime addressing is fine
        float im = res[(2*k + 1)*17 + m];
        float p  = re*re + im*im;
        ev[0] += hi ? -p : p;               // wire 0 sign = bit3 = hi
        ev[1] += ((k2 >> 2) & 1) ? -p : p;  // compile-time signs
        ev[2] += ((k2 >> 1) & 1) ? -p : p;
        ev[3] += ( k2       & 1) ? -p : p;
    }
    #pragma unroll
    for (int i = 0; i < 4; ++i)
        ev[i] += __shfl_xor(ev[i], 16, 32);

    if (hi == 0 && s < B)
        *(float4*)(out + 4 * (size_t)s) = make_float4(ev[0], ev[1], ev[2], ev[3]);
}

// ---------------------------------------------------------------------------
extern "C" void kernel_launch(void* const* d_in, const int* in_sizes, int n_in,
                              void* d_out, int out_size, void* d_ws, size_t ws_size,
                              hipStream_t stream) {
    const float* x  = (const float*)d_in[0];   // [B,4] f32
    const float* qw = (const float*)d_in[1];   // [3,4,3] f32
    float* out = (float*)d_out;                // [B,4] f32
    float* BM  = (float*)d_ws;                 // 32x32 f32 = 4 KB scratch

    int B = in_sizes[0] / 4;

    build_umat<<<1, 32, 0, stream>>>(qw, BM);

    int tiles  = (B + 15) / 16;       // 16 samples per wave
    int blocks = (tiles + 7) / 8;     // 8 waves per 256-thread block
    qlayer_main<<<blocks, 256, 0, stream>>>(x, BM, out, B);
}